// TreeModelAttMDCore_65429531787912
// MI455X (gfx1250) — compile-verified
//
#include <hip/hip_runtime.h>
#include <hip/hip_bf16.h>
#include <math.h>

#define Bn 256
#define Sn 196
#define Hn 1024
#define Fn 2048

typedef float v2f __attribute__((ext_vector_type(2)));
typedef float v8f __attribute__((ext_vector_type(8)));

// ---------------------------------------------------------------------------
// Kernel 1: q[b][n] = sum_k ha[b][k]*W_ha[n][k] + sum_k hf[b][k]*W_hf[n][k]
//                     + b_ha[n] + b_hf[n]
// One wave (32 lanes) computes one 16(M=batch) x 16(N=hidden) fp32 tile using
// V_WMMA_F32_16X16X4_F32, K-loop over 1024 in steps of 4, both weight
// matrices accumulated into the same 8-VGPR accumulator.
//
// ISA per-lane layouts (CDNA5 ISA):
//   A 16x4 f32 : lane -> M = lane&15 ; VGPR0 = K=kb, VGPR1 = K=kb+1,
//                kb = (lane>>4)*2  (0 for lanes 0-15, 2 for lanes 16-31)
//   B 4x16 f32 : lane -> N = lane&15 ; same kb split (rows striped on lanes)
//   C/D 16x16  : VGPR r -> M = r + (lane>>4)*8, N = lane&15
// ---------------------------------------------------------------------------
__global__ __launch_bounds__(128) void qproj_kernel(
    const float* __restrict__ ha, const float* __restrict__ hf,
    const float* __restrict__ W_ha, const float* __restrict__ b_ha,
    const float* __restrict__ W_hf, const float* __restrict__ b_hf,
    float* __restrict__ q) {
  const int lane = threadIdx.x;                       // 0..31
  const int m0 = blockIdx.x * 16;                     // batch tile
  const int n0 = (blockIdx.y * blockDim.y + threadIdx.y) * 16;  // hidden tile
  const int mr = lane & 15;
  const int kb = (lane >> 4) * 2;                     // 0 or 2

  const float* __restrict__ A1 = ha + (size_t)(m0 + mr) * Hn + kb;   // A row (M=mr)
  const float* __restrict__ A2 = hf + (size_t)(m0 + mr) * Hn + kb;
  const float* __restrict__ B1 = W_ha + (size_t)(n0 + mr) * Hn + kb; // B col (N=mr)
  const float* __restrict__ B2 = W_hf + (size_t)(n0 + mr) * Hn + kb;

  v8f acc = {};
#pragma unroll 4
  for (int k = 0; k < Hn; k += 4) {
    v2f a1 = *(const v2f*)(A1 + k);
    v2f b1 = *(const v2f*)(B1 + k);
    acc = __builtin_amdgcn_wmma_f32_16x16x4_f32(
        /*neg_a=*/false, a1, /*neg_b=*/false, b1,
        /*c_mod=*/(short)0, acc, /*reuse_a=*/false, /*reuse_b=*/false);
    v2f a2 = *(const v2f*)(A2 + k);
    v2f b2 = *(const v2f*)(B2 + k);
    acc = __builtin_amdgcn_wmma_f32_16x16x4_f32(
        false, a2, false, b2, (short)0, acc, false, false);
  }

  const int n = n0 + (lane & 15);
  const float bias = b_ha[n] + b_hf[n];
  const int mbase = m0 + (lane >> 4) * 8;
#pragma unroll
  for (int r = 0; r < 8; ++r) {
    q[(size_t)(mbase + r) * Hn + n] = acc[r] + bias;
  }
}

// ---------------------------------------------------------------------------
// Kernel 2: scores[b][s] = sum_h tanh(q[b][h] + p_att[b][s][h]) * w_alpha[h]
//           + b_alpha.  One wave per (b,s) pair; float4 streaming of p_att
//           (the 205 MB stream), wave32 xor-shuffle reduction.
// ---------------------------------------------------------------------------
__device__ __forceinline__ float fast_tanh(float x) {
  float e = __expf(2.0f * x);            // v_exp_f32
  return 1.0f - 2.0f / (e + 1.0f);
}

__global__ __launch_bounds__(256) void score_kernel(
    const float* __restrict__ q, const float* __restrict__ p_att,
    const float* __restrict__ w_alpha, const float* __restrict__ b_alpha,
    float* __restrict__ scores) {
  const int pair = blockIdx.x * blockDim.y + threadIdx.y;  // 0..B*S-1
  const int b = pair / Sn;
  const int s = pair - b * Sn;
  const float* __restrict__ qrow = q + (size_t)b * Hn;
  const float* __restrict__ prow = p_att + ((size_t)b * Sn + s) * Hn;

  float sum = 0.0f;
#pragma unroll
  for (int k = threadIdx.x * 4; k < Hn; k += 32 * 4) {
    float4 pv = *(const float4*)(prow + k);
    float4 qv = *(const float4*)(qrow + k);
    float4 wv = *(const float4*)(w_alpha + k);
    sum = fmaf(fast_tanh(qv.x + pv.x), wv.x, sum);
    sum = fmaf(fast_tanh(qv.y + pv.y), wv.y, sum);
    sum = fmaf(fast_tanh(qv.z + pv.z), wv.z, sum);
    sum = fmaf(fast_tanh(qv.w + pv.w), wv.w, sum);
  }
#pragma unroll
  for (int off = 16; off > 0; off >>= 1)
    sum += __shfl_xor(sum, off, 32);
  if (threadIdx.x == 0) scores[pair] = sum + b_alpha[0];
}

// ---------------------------------------------------------------------------
// Kernel 3: softmax over S=196 per batch row. One block per batch.
// ---------------------------------------------------------------------------
__global__ __launch_bounds__(256) void softmax_kernel(
    const float* __restrict__ scores, float* __restrict__ weight) {
  __shared__ float red[256];
  const int b = blockIdx.x;
  const int t = threadIdx.x;
  float v = (t < Sn) ? scores[b * Sn + t] : -INFINITY;
  red[t] = v;
  __syncthreads();
#pragma unroll
  for (int off = 128; off > 0; off >>= 1) {
    if (t < off) red[t] = fmaxf(red[t], red[t + off]);
    __syncthreads();
  }
  const float mx = red[0];
  __syncthreads();
  float e = (t < Sn) ? __expf(v - mx) : 0.0f;
  red[t] = e;
  __syncthreads();
#pragma unroll
  for (int off = 128; off > 0; off >>= 1) {
    if (t < off) red[t] += red[t + off];
    __syncthreads();
  }
  const float inv = 1.0f / red[0];
  if (t < Sn) weight[b * Sn + t] = e * inv;
}

// ---------------------------------------------------------------------------
// Kernel 4: out[b][f] = sum_s weight[b][s] * att_feats[b][s][f]
// One thread per (b,f); perfectly coalesced streaming of the 411 MB tensor,
// weight[b][s] is block-uniform (scalar-loaded).
// ---------------------------------------------------------------------------
__global__ __launch_bounds__(256) void attres_kernel(
    const float* __restrict__ weight, const float* __restrict__ att_feats,
    float* __restrict__ out) {
  const int f = blockIdx.x * 256 + threadIdx.x;
  const int b = blockIdx.y;
  const float* __restrict__ af = att_feats + (size_t)b * Sn * Fn + f;
  const float* __restrict__ w = weight + (size_t)b * Sn;
  float acc = 0.0f;
#pragma unroll 4
  for (int s = 0; s < Sn; ++s) acc = fmaf(w[s], af[(size_t)s * Fn], acc);
  out[(size_t)b * Fn + f] = acc;
}

// ---------------------------------------------------------------------------
extern "C" void kernel_launch(void* const* d_in, const int* in_sizes, int n_in,
                              void* d_out, int out_size, void* d_ws, size_t ws_size,
                              hipStream_t stream) {
  const float* ha      = (const float*)d_in[0];
  const float* hf      = (const float*)d_in[1];
  const float* att_f   = (const float*)d_in[2];
  const float* p_att   = (const float*)d_in[3];
  const float* W_ha    = (const float*)d_in[4];
  const float* b_ha    = (const float*)d_in[5];
  const float* W_hf    = (const float*)d_in[6];
  const float* b_hf    = (const float*)d_in[7];
  const float* w_alpha = (const float*)d_in[8];
  const float* b_alpha = (const float*)d_in[9];
  float* out = (float*)d_out;

  float* q      = (float*)d_ws;                 // [B,H]  = 1 MB
  float* scores = q + (size_t)Bn * Hn;          // [B,S]
  float* weight = scores + (size_t)Bn * Sn;     // [B,S]

  // 1) fused dual-weight query projection (WMMA fp32)
  dim3 gA(Bn / 16, Hn / 16 / 4), bA(32, 4);
  qproj_kernel<<<gA, bA, 0, stream>>>(ha, hf, W_ha, b_ha, W_hf, b_hf, q);

  // 2) tanh + dot with w_alpha (one wave per (b,s))
  dim3 gB((Bn * Sn) / 8), bB(32, 8);
  score_kernel<<<gB, bB, 0, stream>>>(q, p_att, w_alpha, b_alpha, scores);

  // 3) softmax over S
  softmax_kernel<<<Bn, 256, 0, stream>>>(scores, weight);

  // 4) weighted feature pooling
  dim3 gD(Fn / 256, Bn);
  attres_kernel<<<gD, 256, 0, stream>>>(weight, att_f, out);
}